// YoloLayer_9955734192643
// MI455X (gfx1250) — compile-verified
//
#include <hip/hip_runtime.h>
#include <stdint.h>

#define NCLS        80
#define N_TOTAL     16128
#define N_PAD       16384
#define PRE_K       400
#define MAX_BOXES   200
#define FLAT_N      32000
#define FLAT_PAD    32768
#define NMS_TH      0.6f
#define NW          13          // 400 bits -> 13 u32 words

typedef unsigned long long u64;
typedef float v2f __attribute__((ext_vector_type(2)));
typedef float v8f __attribute__((ext_vector_type(8)));

// ---------------- helpers ----------------
__device__ __forceinline__ float sigmoidf_(float x) { return 1.0f / (1.0f + __expf(-x)); }

// monotonic float -> u32 (total order), real scores are positive, pads are -1
__device__ __forceinline__ unsigned fkey(float f) {
    unsigned u = __float_as_uint(f);
    return (u & 0x80000000u) ? ~u : (u | 0x80000000u);
}
__device__ __forceinline__ float fdekey(unsigned k) {
    return (k & 0x80000000u) ? __uint_as_float(k & 0x7FFFFFFFu) : __uint_as_float(~k);
}
__device__ __forceinline__ u64 mkkey(float s, unsigned idx) {
    return ((u64)fkey(s) << 32) | (u64)(0xFFFFFFFFu - idx);   // desc score, asc index on ties
}

// generic shared pointer -> wave-relative LDS byte offset (flat addr low 32 bits)
__device__ __forceinline__ unsigned lds_off(const void* p) { return (unsigned)(uintptr_t)p; }

// 512-element descending bitonic sort in LDS, 256 threads
__device__ void sort512_desc(u64* key, int tid) {
    for (unsigned k = 2; k <= 512; k <<= 1)
        for (unsigned j = k >> 1; j > 0; j >>= 1) {
            __syncthreads();
            #pragma unroll 1
            for (unsigned i = tid; i < 512; i += 256) {
                unsigned ixj = i ^ j;
                if (ixj > i) {
                    u64 a = key[i], b = key[ixj];
                    bool desc = ((i & k) == 0);
                    if (desc ? (a < b) : (a > b)) { key[i] = b; key[ixj] = a; }
                }
            }
        }
    __syncthreads();
}

// A desc(512), B desc(512) -> A = top-512 of union, sorted desc
__device__ void merge_top512(u64* A, const u64* B, int tid) {
    __syncthreads();
    for (unsigned i = tid; i < 512; i += 256) {
        u64 a = A[i], b = B[511 - i];
        if (b > a) A[i] = b;                 // keep max half; A is now bitonic
    }
    for (unsigned j = 256; j > 0; j >>= 1) {
        __syncthreads();
        for (unsigned i = tid; i < 512; i += 256) {
            unsigned ixj = i ^ j;
            if (ixj > i) {
                u64 a = A[i], b = A[ixj];
                if (a < b) { A[i] = b; A[ixj] = a; }
            }
        }
    }
    __syncthreads();
}

// ---------------- kernels ----------------
__global__ void fill_f32(float* p, int n, float v) {
    int i = blockIdx.x * blockDim.x + threadIdx.x;
    if (i < n) p[i] = v;
}

// one thread per (b,h,w,anchor): decode box + 80 class scores
__global__ void decode_kernel(const float* __restrict__ p, const float* __restrict__ anchors,
                              float* __restrict__ scoresT, float4* __restrict__ boxes,
                              int B, int H, int W, float sxy, float stride, int nOffset) {
    int idx = blockIdx.x * blockDim.x + threadIdx.x;
    int total = B * H * W * 3;
    if (idx >= total) return;
    int a = idx % 3;
    int w = (idx / 3) % W;
    int h = (idx / (3 * W)) % H;
    int b = idx / (3 * W * H);

    const float* cell = p + (((size_t)b * H + h) * W + w) * (3 * (NCLS + 5)) + (size_t)a * (NCLS + 5);
    float tx = cell[0], ty = cell[1], tw = cell[2], th = cell[3];
    float sobj = sigmoidf_(cell[4]);

    float half_s = 0.5f * (sxy - 1.0f);
    float cx = (sigmoidf_(tx) * sxy - half_s + (float)w) / (float)W;
    float cy = (sigmoidf_(ty) * sxy - half_s + (float)h) / (float)H;
    float bw = __expf(tw) * anchors[a * 2 + 0] / ((float)W * stride);
    float bh = __expf(th) * anchors[a * 2 + 1] / ((float)H * stride);

    float y1 = fminf(fmaxf(cy - 0.5f * bh, 0.0f), 1.0f);
    float x1 = fminf(fmaxf(cx - 0.5f * bw, 0.0f), 1.0f);
    float y2 = fminf(fmaxf(cy + 0.5f * bh, 0.0f), 1.0f);
    float x2 = fminf(fmaxf(cx + 0.5f * bw, 0.0f), 1.0f);

    int n = nOffset + (h * W + w) * 3 + a;
    boxes[(size_t)b * N_TOTAL + n] = make_float4(y1, x1, y2, x2);

    float* srow = scoresT + (size_t)b * NCLS * N_PAD;
    #pragma unroll 4
    for (int c = 0; c < NCLS; ++c)
        srow[(size_t)c * N_PAD + n] = sigmoidf_(cell[5 + c]) * sobj;
}

// one block per (image,class): top-400 -> WMMA IoU tiles -> greedy NMS -> per-class slots
__global__ void __launch_bounds__(256)
nms_class_kernel(const float* __restrict__ scoresT, const float4* __restrict__ boxes,
                 float* __restrict__ flatScores, float4* __restrict__ flatBoxes) {
    __shared__ u64 keyA[512];
    __shared__ u64 keyB[512];
    __shared__ float    topScore[PRE_K];
    __shared__ unsigned topIdx[PRE_K];
    __shared__ float4   boxL[PRE_K];
    __shared__ float    areaL[PRE_K];
    __shared__ unsigned ov[PRE_K * NW];
    __shared__ unsigned keepW[NW];

    const int tid = threadIdx.x;
    const int b = blockIdx.x / NCLS;
    const int c = blockIdx.x % NCLS;
    const float* srow = scoresT + ((size_t)b * NCLS + c) * N_PAD;

    // ---- phase A: streaming top-512 over 16384 padded scores ----
    for (unsigned i = tid; i < 512; i += 256) keyA[i] = mkkey(srow[i], i);
    sort512_desc(keyA, tid);
    for (int ch = 1; ch < N_PAD / 512; ++ch) {
        __syncthreads();
        for (unsigned i = tid; i < 512; i += 256) {
            unsigned e = ch * 512 + i;
            keyB[i] = mkkey(srow[e], e);
        }
        sort512_desc(keyB, tid);
        merge_top512(keyA, keyB, tid);
    }
    for (unsigned i = tid; i < PRE_K; i += 256) {
        u64 kk = keyA[i];
        topScore[i] = fdekey((unsigned)(kk >> 32));
        unsigned id = 0xFFFFFFFFu - (unsigned)(kk & 0xFFFFFFFFu);
        topIdx[i] = (id < N_TOTAL) ? id : 0u;
    }
    __syncthreads();

    // ---- phase B: async gather of 400 boxes into LDS (ASYNCcnt path) ----
    for (unsigned i = tid; i < PRE_K; i += 256) {
        const float4* g = boxes + (size_t)b * N_TOTAL + topIdx[i];
        unsigned dst = lds_off(&boxL[i]);
        asm volatile("global_load_async_to_lds_b128 %0, %1, off"
                     :: "v"(dst), "v"((u64)(uintptr_t)g) : "memory");
    }
    asm volatile("s_wait_asynccnt 0x0" ::: "memory");
    __syncthreads();

    for (unsigned i = tid; i < PRE_K; i += 256) {
        float4 bx = boxL[i];
        areaL[i] = (bx.z - bx.x) * (bx.w - bx.y);
    }
    for (unsigned i = tid; i < PRE_K * NW; i += 256) ov[i] = 0u;
    __syncthreads();

    // ---- phase C: 625 16x16 IoU tiles; union base (area_i + area_j) via WMMA ----
    const int lane = tid & 31, wv = tid >> 5;
    const int half = lane >> 4, m = lane & 15;
    for (int job = wv; job < 25 * 25; job += 8) {
        int ti = job / 25, tj = job - ti * 25;
        v2f A, Bv;
        if (half == 0) { A.x = areaL[ti * 16 + m]; A.y = 1.0f;
                         Bv.x = 1.0f;             Bv.y = areaL[tj * 16 + m]; }
        else           { A.x = 0.0f; A.y = 0.0f; Bv.x = 0.0f; Bv.y = 0.0f; }
        v8f cz = {0.f,0.f,0.f,0.f,0.f,0.f,0.f,0.f};
        v8f d = __builtin_amdgcn_wmma_f32_16x16x4_f32(false, A, false, Bv,
                                                      (short)0, cz, false, false);
        #pragma unroll
        for (int r = 0; r < 8; ++r) {
            int i = ti * 16 + r + 8 * half;    // D layout: lanes16-31 hold M=r+8
            int j = tj * 16 + m;
            float4 bi = boxL[i], bj = boxL[j];
            float iw = fmaxf(fminf(bi.z, bj.z) - fmaxf(bi.x, bj.x), 0.0f);
            float ih = fmaxf(fminf(bi.w, bj.w) - fmaxf(bi.y, bj.y), 0.0f);
            float inter = iw * ih;
            float uni = fmaxf(d[r] - inter, 1e-9f);
            if (inter > NMS_TH * uni)
                atomicOr(&ov[i * NW + (j >> 5)], 1u << (j & 31));
        }
    }
    __syncthreads();

    // ---- phase D: serial greedy suppression over the bitmask ----
    if (tid < NW) keepW[tid] = (tid < 12) ? 0xFFFFFFFFu : 0x0000FFFFu;
    __syncthreads();
    for (int i = 0; i < PRE_K; ++i) {
        bool ki = (keepW[i >> 5] >> (i & 31)) & 1u;
        if (ki && tid < NW) {
            unsigned mrow = ov[i * NW + tid];
            int w = i >> 5, r = i & 31;
            if (tid < w) mrow = 0u;
            else if (tid == w) mrow &= (r == 31) ? 0u : ~((1u << (r + 1)) - 1u); // only j > i
            keepW[tid] &= ~mrow;
        }
        __syncthreads();
    }

    // ---- phase E: emit per-class slots ----
    float*  fs = flatScores + (size_t)b * FLAT_PAD + (size_t)c * PRE_K;
    float4* fb = flatBoxes  + ((size_t)b * NCLS + c) * PRE_K;
    for (unsigned i = tid; i < PRE_K; i += 256) {
        bool kp = (keepW[i >> 5] >> (i & 31)) & 1u;
        float s = topScore[i];
        fs[i] = (kp && s > 0.0f) ? s : 0.0f;
        fb[i] = boxL[i];
    }
}

// one block per image: top-200 of 32000 + final outputs
__global__ void __launch_bounds__(256)
final_topk_kernel(const float* __restrict__ flatScores, const float4* __restrict__ flatBoxes,
                  float* __restrict__ out) {
    __shared__ u64 keyA[512];
    __shared__ u64 keyB[512];
    __shared__ int cnt;

    const int tid = threadIdx.x;
    const int b = blockIdx.x;
    const float* row = flatScores + (size_t)b * FLAT_PAD;

    for (unsigned i = tid; i < 512; i += 256) keyA[i] = mkkey(row[i], i);
    sort512_desc(keyA, tid);
    for (int ch = 1; ch < FLAT_PAD / 512; ++ch) {
        __syncthreads();
        for (unsigned i = tid; i < 512; i += 256) {
            unsigned e = ch * 512 + i;
            keyB[i] = mkkey(row[e], e);
        }
        sort512_desc(keyB, tid);
        merge_top512(keyA, keyB, tid);
    }

    if (tid == 0) cnt = 0;
    __syncthreads();

    float* bboxO = out;                       // [8][200][4]
    float* confO = out + 8 * MAX_BOXES * 4;   // [8][200]
    float* clsO  = out + 8 * MAX_BOXES * 5;   // [8][200]
    float* numO  = out + 8 * MAX_BOXES * 6;   // [8]

    for (unsigned i = tid; i < MAX_BOXES; i += 256) {
        u64 kk = keyA[i];
        float s = fdekey((unsigned)(kk >> 32));
        unsigned f = 0xFFFFFFFFu - (unsigned)(kk & 0xFFFFFFFFu);
        bool valid = (s > 0.0f) && (f < FLAT_N);
        float4 bx = valid ? flatBoxes[(size_t)b * FLAT_N + f]
                          : make_float4(0.f, 0.f, 0.f, 0.f);
        size_t o = (size_t)b * MAX_BOXES + i;
        bboxO[o * 4 + 0] = bx.x;  bboxO[o * 4 + 1] = bx.y;
        bboxO[o * 4 + 2] = bx.z;  bboxO[o * 4 + 3] = bx.w;
        confO[o] = valid ? s : 0.0f;
        clsO[o]  = valid ? (float)(f / PRE_K) : 0.0f;
        if (valid) atomicAdd(&cnt, 1);
    }
    __syncthreads();
    if (tid == 0) numO[b] = (float)cnt;
}

// ---------------- host ----------------
extern "C" void kernel_launch(void* const* d_in, const int* in_sizes, int n_in,
                              void* d_out, int out_size, void* d_ws, size_t ws_size,
                              hipStream_t stream) {
    (void)in_sizes; (void)n_in; (void)out_size; (void)ws_size;
    const float* p3 = (const float*)d_in[0];
    const float* p4 = (const float*)d_in[1];
    const float* p5 = (const float*)d_in[2];
    const float* a3 = (const float*)d_in[3];
    const float* a4 = (const float*)d_in[4];
    const float* a5 = (const float*)d_in[5];
    float* out = (float*)d_out;

    const int B = 8;
    float*  scoresT    = (float*)d_ws;                                    // 8*80*16384
    float4* boxesWS    = (float4*)(scoresT + (size_t)B * NCLS * N_PAD);   // 8*16128
    float*  flatScores = (float*)(boxesWS + (size_t)B * N_TOTAL);         // 8*32768
    float4* flatBoxes  = (float4*)(flatScores + (size_t)B * FLAT_PAD);    // 8*32000

    int nS = B * NCLS * N_PAD;
    fill_f32<<<(nS + 255) / 256, 256, 0, stream>>>(scoresT, nS, -1.0f);
    int nF = B * FLAT_PAD;
    fill_f32<<<(nF + 255) / 256, 256, 0, stream>>>(flatScores, nF, -1.0f);

    int t3 = B * 64 * 64 * 3, t4 = B * 32 * 32 * 3, t5 = B * 16 * 16 * 3;
    decode_kernel<<<(t3 + 255) / 256, 256, 0, stream>>>(p3, a3, scoresT, boxesWS,
                                                        B, 64, 64, 1.20f, 8.0f, 0);
    decode_kernel<<<(t4 + 255) / 256, 256, 0, stream>>>(p4, a4, scoresT, boxesWS,
                                                        B, 32, 32, 1.10f, 16.0f, 12288);
    decode_kernel<<<(t5 + 255) / 256, 256, 0, stream>>>(p5, a5, scoresT, boxesWS,
                                                        B, 16, 16, 1.05f, 32.0f, 15360);

    nms_class_kernel<<<B * NCLS, 256, 0, stream>>>(scoresT, boxesWS, flatScores, flatBoxes);
    final_topk_kernel<<<B, 256, 0, stream>>>(flatScores, flatBoxes, out);
}